// ReadoutModule2_79345225826307
// MI455X (gfx1250) — compile-verified
//
#include <hip/hip_runtime.h>
#include <hip/hip_bf16.h>
#include <math.h>

// Problem constants (from reference)
#define KHID 256     // feature width
#define KSEG 2048    // number of segments (B)

typedef float v2f __attribute__((ext_vector_type(2)));
typedef float v8f __attribute__((ext_vector_type(8)));

// ---------------------------------------------------------------------------
// Kernel 0: zero-fill (d_out and workspace accumulators must start at 0)
// ---------------------------------------------------------------------------
__global__ void zero_kernel(float* __restrict__ p, int n) {
    int i = blockIdx.x * blockDim.x + threadIdx.x;
    int stride = gridDim.x * blockDim.x;
    for (; i < n; i += stride) p[i] = 0.0f;
}

// ---------------------------------------------------------------------------
// Kernel 1: segment sums + counts.
// One wave per contiguous chunk of rows. Each lane owns 8 feature columns
// (lane*8 .. lane*8+7) -> two float4 loads per row, fully coalesced (1KB/row
// per wave). batch is sorted, so we keep a running register accumulator and
// flush with global_atomic_add_f32 only on segment boundaries.
// ---------------------------------------------------------------------------
__global__ void pool_kernel(const float* __restrict__ x,
                            const int*   __restrict__ batch,
                            float* __restrict__ sums,
                            float* __restrict__ counts,
                            int nrows, int rows_per_wave) {
    const int wave = (blockIdx.x * blockDim.x + threadIdx.x) >> 5;
    const int lane = threadIdx.x & 31;
    int r0 = wave * rows_per_wave;
    int r1 = min(nrows, r0 + rows_per_wave);
    if (r0 >= r1) return;

    const float4* x4 = (const float4*)x;
    float acc[8] = {0.f,0.f,0.f,0.f,0.f,0.f,0.f,0.f};
    float cnt = 0.0f;
    int cur = batch[r0];

    for (int r = r0; r < r1; ++r) {
        int seg = batch[r];
        if (seg != cur) {
            float* dst = sums + (size_t)cur * KHID + lane * 8;
            #pragma unroll
            for (int j = 0; j < 8; ++j) atomicAdd(dst + j, acc[j]);
            if (lane == 0) atomicAdd(counts + cur, cnt);
            #pragma unroll
            for (int j = 0; j < 8; ++j) acc[j] = 0.0f;
            cnt = 0.0f;
            cur = seg;
        }
        size_t base = (size_t)r * (KHID / 4) + lane * 2;
        float4 u = x4[base];
        float4 v = x4[base + 1];
        acc[0] += u.x; acc[1] += u.y; acc[2] += u.z; acc[3] += u.w;
        acc[4] += v.x; acc[5] += v.y; acc[6] += v.z; acc[7] += v.w;
        cnt += 1.0f;
    }
    float* dst = sums + (size_t)cur * KHID + lane * 8;
    #pragma unroll
    for (int j = 0; j < 8; ++j) atomicAdd(dst + j, acc[j]);
    if (lane == 0) atomicAdd(counts + cur, cnt);
}

// ---------------------------------------------------------------------------
// Kernel 2: tg = tanh( (sums / max(counts,1)) @ weight )  via V_WMMA_F32_16X16X4_F32
// One wave computes one 16x16 output tile; K=256 stepped by 4 => 64 WMMAs/wave.
// 2048 tiles total (128 M-tiles x 16 N-tiles) = 2048 waves.
// Fragment layouts per CDNA5 ISA 7.12.2 (32-bit A 16x4 / B 4x16 / C 16x16).
// ---------------------------------------------------------------------------
__global__ void gemm_tanh_kernel(const float* __restrict__ sums,
                                 const float* __restrict__ counts,
                                 const float* __restrict__ weight,
                                 float* __restrict__ tg) {
    const int wave   = (blockIdx.x * blockDim.x + threadIdx.x) >> 5;
    const int lane   = threadIdx.x & 31;
    const int tile_m = wave >> 4;        // 0..127
    const int tile_n = wave & 15;        // 0..15
    const int hi     = lane >> 4;        // 0: K+{0,1}, 1: K+{2,3}
    const int l      = lane & 15;

    const int m = tile_m * 16 + l;       // A-row owned by this lane
    const int n = tile_n * 16 + l;       // B-col owned by this lane
    const float inv_cnt = 1.0f / fmaxf(counts[m], 1.0f);

    v8f c = {};
    for (int k0 = 0; k0 < KHID; k0 += 4) {
        const int ka = k0 + 2 * hi;
        v2f a, b;
        a[0] = sums[(size_t)m * KHID + ka]     * inv_cnt;   // A[m][ka]
        a[1] = sums[(size_t)m * KHID + ka + 1] * inv_cnt;   // A[m][ka+1]
        b[0] = weight[(size_t)(ka)     * KHID + n];         // B[ka][n]
        b[1] = weight[(size_t)(ka + 1) * KHID + n];         // B[ka+1][n]
        c = __builtin_amdgcn_wmma_f32_16x16x4_f32(
                /*neg_a=*/false, a, /*neg_b=*/false, b,
                /*c_mod=*/(short)0, c, /*reuse_a=*/false, /*reuse_b=*/false);
    }
    #pragma unroll
    for (int v = 0; v < 8; ++v) {
        int row = tile_m * 16 + v + 8 * hi;  // C layout: VGPR v -> M=v (lo) / v+8 (hi)
        int col = tile_n * 16 + l;
        tg[(size_t)row * KHID + col] = tanhf(c[v]);
    }
}

// ---------------------------------------------------------------------------
// Kernel 3: coefs + weighted segment-sum.
// Same wave-per-chunk structure: per row, lane-local 8-wide partial dot,
// wave-wide xor-shuffle reduction (wave32), sigmoid, accumulate c*x into a
// register tile, flush with atomics on segment boundary.
// ---------------------------------------------------------------------------
__global__ void gate_kernel(const float* __restrict__ x,
                            const int*   __restrict__ batch,
                            const float* __restrict__ tg,
                            float* __restrict__ out,
                            int nrows, int rows_per_wave) {
    const int wave = (blockIdx.x * blockDim.x + threadIdx.x) >> 5;
    const int lane = threadIdx.x & 31;
    int r0 = wave * rows_per_wave;
    int r1 = min(nrows, r0 + rows_per_wave);
    if (r0 >= r1) return;

    const float4* x4 = (const float4*)x;
    float acc[8] = {0.f,0.f,0.f,0.f,0.f,0.f,0.f,0.f};
    float tgv[8];
    int cur = batch[r0];
    {
        const float* t = tg + (size_t)cur * KHID + lane * 8;
        #pragma unroll
        for (int j = 0; j < 8; ++j) tgv[j] = t[j];
    }

    for (int r = r0; r < r1; ++r) {
        int seg = batch[r];
        if (seg != cur) {
            float* dst = out + (size_t)cur * KHID + lane * 8;
            #pragma unroll
            for (int j = 0; j < 8; ++j) atomicAdd(dst + j, acc[j]);
            #pragma unroll
            for (int j = 0; j < 8; ++j) acc[j] = 0.0f;
            cur = seg;
            const float* t = tg + (size_t)cur * KHID + lane * 8;
            #pragma unroll
            for (int j = 0; j < 8; ++j) tgv[j] = t[j];
        }
        size_t base = (size_t)r * (KHID / 4) + lane * 2;
        float4 u = x4[base];
        float4 v = x4[base + 1];
        float xv[8] = {u.x, u.y, u.z, u.w, v.x, v.y, v.z, v.w};

        float p = 0.0f;
        #pragma unroll
        for (int j = 0; j < 8; ++j) p = fmaf(xv[j], tgv[j], p);
        // wave32 xor-reduction: every lane ends with the full 256-wide dot
        #pragma unroll
        for (int off = 16; off > 0; off >>= 1) p += __shfl_xor(p, off, 32);

        float cgate = 1.0f / (1.0f + expf(-p));
        #pragma unroll
        for (int j = 0; j < 8; ++j) acc[j] = fmaf(cgate, xv[j], acc[j]);
    }
    float* dst = out + (size_t)cur * KHID + lane * 8;
    #pragma unroll
    for (int j = 0; j < 8; ++j) atomicAdd(dst + j, acc[j]);
}

// ---------------------------------------------------------------------------
extern "C" void kernel_launch(void* const* d_in, const int* in_sizes, int n_in,
                              void* d_out, int out_size, void* d_ws, size_t ws_size,
                              hipStream_t stream) {
    const float* x      = (const float*)d_in[0];   // [N, 256] fp32
    const int*   batch  = (const int*)  d_in[1];   // [N] sorted segment ids
    const float* weight = (const float*)d_in[2];   // [256, 256] fp32
    float*       out    = (float*)d_out;           // [2048, 256] fp32

    const int nrows = in_sizes[0] / KHID;          // N = 500000

    // Workspace layout: sums[2048*256] | counts[2048] | tg[2048*256]
    float* sums   = (float*)d_ws;
    float* counts = sums + (size_t)KSEG * KHID;
    float* tg     = counts + KSEG;

    // 0) zero accumulators + output (harness poisons them with 0xAA)
    zero_kernel<<<512, 256, 0, stream>>>(sums, KSEG * KHID + KSEG);
    zero_kernel<<<512, 256, 0, stream>>>(out, KSEG * KHID);

    // 1) mean-pool accumulation: 512 blocks x 8 waves = 4096 waves
    const int nwaves = 512 * (256 / 32);
    const int rpw = (nrows + nwaves - 1) / nwaves;
    pool_kernel<<<512, 256, 0, stream>>>(x, batch, sums, counts, nrows, rpw);

    // 2) tg = tanh(mean_pool @ weight): 2048 tiles, 8 waves/block -> 256 blocks
    gemm_tanh_kernel<<<256, 256, 0, stream>>>(sums, counts, weight, tg);

    // 3) gated weighted segment sum
    gate_kernel<<<512, 256, 0, stream>>>(x, batch, tg, out, nrows, rpw);
}